// GDE_73229192397035
// MI455X (gfx1250) — compile-verified
//
#include <hip/hip_runtime.h>
#include <hip/hip_bf16.h>

// ---------------------------------------------------------------------------
// MI455X (gfx1250) implementation.
//
// Roofline: 5.8 GFLOP vs ~370 MB of once-read data (gathered L rows + masks)
// -> HBM bound (~16us floor @ 23.3 TB/s). Keep f32 end-to-end and use the
// f32 matrix pipe: v_wmma_f32_16x16x4_f32 (16x16 f32 C/D, K=4).
//
// Phase 1: 1536 single-wave workgroups; each computes a 16(batch) x 64(D)
//          partial slab over a K-chunk with 4 WMMAs per K-step.
//          L/mask streams are non-temporal (read-once, keep embed in L2);
//          embed tables (<=4 MB) stay L2-resident and are reused by all waves.
// Phase 2: per-batch-tile deterministic reduction over chunks + dot products
//          + sigmoid/log epilogue + regularizer partials.
// Phase 3: single-wave final reduction -> d_out[0].
//
// Workspace: 3*64*8*1024 floats (6,291,456 B) partials + 64 floats tile sums.
// ---------------------------------------------------------------------------

typedef __attribute__((ext_vector_type(2))) float v2f;
typedef __attribute__((ext_vector_type(8))) float v8f;

#define NUx 12000
#define NIx 16000
#define DD 64
#define BBx 1024
#define NCHUNK 8
#define BTILES (BBx / 16)   // 64
#define DROPP 0.1f
#define REGC 0.01f

__global__ __launch_bounds__(32) void gde_gemm_partial(
    const float* __restrict__ Lu, const float* __restrict__ Li,
    const float* __restrict__ Eu, const float* __restrict__ Ei,
    const float* __restrict__ Mu, const float* __restrict__ Mp,
    const float* __restrict__ Mn,
    const int* __restrict__ users, const int* __restrict__ pos,
    const int* __restrict__ neg,
    float* __restrict__ Part)
{
    const int lane  = threadIdx.x;   // one wave32 per block
    const int mrow  = lane & 15;     // M index 0..15
    const int khalf = lane >> 4;     // 0: K=0,1  1: K=2,3

    const int t     = blockIdx.x;
    const int g     = t / (BTILES * NCHUNK);          // 0=user 1=pos 2=neg
    const int r     = t - g * (BTILES * NCHUNK);
    const int btile = r / NCHUNK;
    const int chunk = r - btile * NCHUNK;

    const float* L; const float* E; const float* M; const int* idx; int N;
    if (g == 0)      { L = Lu; E = Eu; M = Mu; idx = users; N = NUx; }
    else if (g == 1) { L = Li; E = Ei; M = Mp; idx = pos;   N = NIx; }
    else             { L = Li; E = Ei; M = Mn; idx = neg;   N = NIx; }

    const int b   = btile * 16 + mrow;
    const int row = idx[b];
    const float* __restrict__ Lrow = L + (size_t)row * (size_t)N;
    const float* __restrict__ Mrow = M + (size_t)b   * (size_t)N;

    const int clen = N / NCHUNK;     // 1500 or 2000 (multiple of 4)
    const int k0   = chunk * clen;
    const int k1   = k0 + clen;

    v8f acc0 = {}, acc1 = {}, acc2 = {}, acc3 = {};

    #pragma unroll 4
    for (int k = k0; k < k1; k += 4) {
        const int ka = k + 2 * khalf;

        // A tile 16x4: gathered L row element pair, zeroed by dropout mask.
        // Read-once streams -> non-temporal (TH=NT): don't churn the 192MB L2.
        v2f lv = __builtin_nontemporal_load((const v2f*)(Lrow + ka));
        v2f mv = __builtin_nontemporal_load((const v2f*)(Mrow + ka));
        v2f a;
        a.x = (mv.x >= DROPP) ? lv.x : 0.0f;
        a.y = (mv.y >= DROPP) ? lv.y : 0.0f;

        // Four B tiles 4x16 covering d = 0..63 (embed rows ka, ka+1).
        // Embed tables are tiny and heavily reused -> regular temporal loads.
        const float* __restrict__ e0 = E + (size_t)ka * DD + mrow;
        v2f b0, b1, b2, b3;
        b0.x = e0[0];  b0.y = e0[DD];
        b1.x = e0[16]; b1.y = e0[DD + 16];
        b2.x = e0[32]; b2.y = e0[DD + 32];
        b3.x = e0[48]; b3.y = e0[DD + 48];

        acc0 = __builtin_amdgcn_wmma_f32_16x16x4_f32(false, a, false, b0, (short)0, acc0, false, false);
        acc1 = __builtin_amdgcn_wmma_f32_16x16x4_f32(false, a, false, b1, (short)0, acc1, false, false);
        acc2 = __builtin_amdgcn_wmma_f32_16x16x4_f32(false, a, false, b2, (short)0, acc2, false, false);
        acc3 = __builtin_amdgcn_wmma_f32_16x16x4_f32(false, a, false, b3, (short)0, acc3, false, false);
    }

    // C/D layout: VGPR rr -> M = rr + 8*khalf_of_lane, N = lane&15.
    float* __restrict__ pd =
        Part + (((size_t)g * BTILES + btile) * NCHUNK + chunk) * (16 * DD);
    const int mbase = khalf * 8;
    #pragma unroll
    for (int rr = 0; rr < 8; ++rr) {
        pd[(mbase + rr) * DD + mrow +  0] = acc0[rr];
        pd[(mbase + rr) * DD + mrow + 16] = acc1[rr];
        pd[(mbase + rr) * DD + mrow + 32] = acc2[rr];
        pd[(mbase + rr) * DD + mrow + 48] = acc3[rr];
    }
}

__global__ __launch_bounds__(32) void gde_reduce_tile(
    const float* __restrict__ Part, float* __restrict__ TileP)
{
    const int lane  = threadIdx.x;   // 0..31, handles d=lane and d=lane+32
    const int btile = blockIdx.x;    // 0..63

    const size_t gstride = (size_t)BTILES * NCHUNK * 16 * DD;
    const size_t tstride = (size_t)NCHUNK * 16 * DD;
    const float* __restrict__ Pu = Part +               (size_t)btile * tstride;
    const float* __restrict__ Pp = Part + gstride     + (size_t)btile * tstride;
    const float* __restrict__ Pn = Part + 2 * gstride + (size_t)btile * tstride;

    float total = 0.0f;
    for (int b = 0; b < 16; ++b) {
        const int o0 = b * DD + lane;
        const int o1 = o0 + 32;
        float fu0 = 0.f, fu1 = 0.f, fp0 = 0.f, fp1 = 0.f, fn0 = 0.f, fn1 = 0.f;
        #pragma unroll
        for (int c = 0; c < NCHUNK; ++c) {
            const int co = c * (16 * DD);
            fu0 += Pu[co + o0]; fu1 += Pu[co + o1];
            fp0 += Pp[co + o0]; fp1 += Pp[co + o1];
            fn0 += Pn[co + o0]; fn1 += Pn[co + o1];
        }
        float dp = fu0 * fp0 + fu1 * fp1;                 // <fu, fp>
        float dn = fu0 * fn0 + fu1 * fn1;                 // <fu, fn>
        float sq = fu0*fu0 + fu1*fu1 + fp0*fp0 + fp1*fp1 + fn0*fn0 + fn1*fn1;
        #pragma unroll
        for (int off = 16; off > 0; off >>= 1) {
            dp += __shfl_xor(dp, off, 32);
            dn += __shfl_xor(dn, off, 32);
            sq += __shfl_xor(sq, off, 32);
        }
        const float s  = 1.0f / (1.0f + expf(-dn));
        const float nw = 1.0f - log10f(1.0f - fminf(s, 0.99f));
        const float o  = 1.0f / (1.0f + expf(-(dp - nw * dn)));
        total += -logf(o) + REGC * sq;
    }
    if (lane == 0) TileP[btile] = total;
}

__global__ __launch_bounds__(32) void gde_final(
    const float* __restrict__ TileP, float* __restrict__ out)
{
    const int lane = threadIdx.x;
    float v = TileP[lane] + TileP[lane + 32];
    #pragma unroll
    for (int off = 16; off > 0; off >>= 1) v += __shfl_xor(v, off, 32);
    if (lane == 0) out[0] = v / (float)BBx;
}

extern "C" void kernel_launch(void* const* d_in, const int* in_sizes, int n_in,
                              void* d_out, int out_size, void* d_ws, size_t ws_size,
                              hipStream_t stream) {
    const float* Lu = (const float*)d_in[0];
    const float* Li = (const float*)d_in[1];
    const float* Eu = (const float*)d_in[2];
    const float* Ei = (const float*)d_in[3];
    const float* Mu = (const float*)d_in[4];
    const float* Mp = (const float*)d_in[5];
    const float* Mn = (const float*)d_in[6];
    const int* users = (const int*)d_in[7];
    const int* pos   = (const int*)d_in[8];
    const int* neg   = (const int*)d_in[9];

    float* Part  = (float*)d_ws;                                   // 6,291,456 B
    float* TileP = Part + (size_t)3 * BTILES * NCHUNK * 16 * DD;   // +64 floats

    hipLaunchKernelGGL(gde_gemm_partial, dim3(3 * BTILES * NCHUNK), dim3(32), 0,
                       stream, Lu, Li, Eu, Ei, Mu, Mp, Mn, users, pos, neg, Part);
    hipLaunchKernelGGL(gde_reduce_tile, dim3(BTILES), dim3(32), 0, stream,
                       Part, TileP);
    hipLaunchKernelGGL(gde_final, dim3(1), dim3(32), 0, stream,
                       TileP, (float*)d_out);
}